// BasicGNN_24240795418940
// MI455X (gfx1250) — compile-verified
//
#include <hip/hip_runtime.h>
#include <hip/hip_bf16.h>

#define N_NODES 100000
#define N_EDGES 1600000
#define D 128
#define XS_STRIDE 132   // pad: row stride mod 64 banks = 4 -> conflict-free A reads; 16B aligned rows

typedef float v2f __attribute__((ext_vector_type(2)));
typedef float v8f __attribute__((ext_vector_type(8)));

// ---------------------------------------------------------------- degree prep
__global__ __launch_bounds__(256) void k_zero(float* __restrict__ deg) {
    int i = blockIdx.x * 256 + threadIdx.x;
    if (i < N_NODES) deg[i] = 0.0f;
}

__global__ __launch_bounds__(256) void k_degree(const long long* __restrict__ ei,
                                                float* __restrict__ deg) {
    int e = blockIdx.x * 256 + threadIdx.x;
    if (e < N_EDGES) atomicAdd(&deg[(int)ei[e]], 1.0f);  // rows are ei[0][:]
}

__global__ __launch_bounds__(256) void k_rsqrt(const float* __restrict__ deg,
                                               float* __restrict__ dinv) {
    int i = blockIdx.x * 256 + threadIdx.x;
    if (i < N_NODES) dinv[i] = rsqrtf(deg[i]);  // deg==0 -> inf, same as deg**-0.5
}

// ------------------------------------------------- fused dual GEMM via WMMA f32
// One block (8 waves) owns one 16-row x tile; wave w owns output cols [16w,16w+16).
// x tile staged once in LDS (coalesced float4 loads), A-frags served by ds_load.
// A-frag (16x4 f32, 2 VGPR): lanes 0-15: M=lane, K=k+{0,1}; lanes 16-31: M=lane-16, K=k+{2,3}
// C/D (16x16 f32, 8 VGPR):   VGPR v: M = v + 8*(lane>=16), N = lane%16
__global__ __launch_bounds__(256) void k_gemm_fused(const float* __restrict__ x,
                                                    const float* __restrict__ Ws,
                                                    const float* __restrict__ Wn,
                                                    const float* __restrict__ bias,
                                                    float* __restrict__ out,
                                                    float* __restrict__ h) {
    __shared__ float xs[16 * XS_STRIDE];

    const int wave = threadIdx.x >> 5;
    const int lane = threadIdx.x & 31;
    const int m0   = blockIdx.x * 16;     // node-row tile (6250 blocks, exact)
    const int n0   = wave * 16;           // output-col tile
    const int lr   = lane & 15;
    const int hi   = lane >> 4;           // 0: K pair {0,1}; 1: K pair {2,3}

    // ---- stage 16x128 x tile into LDS: 512 float4, 2 per thread, coalesced
    {
        const float4* src = (const float4*)(x + (long long)m0 * D);
        #pragma unroll
        for (int i = 0; i < 2; ++i) {
            const int idx = threadIdx.x + i * 256;  // float4 index 0..511
            const int row = idx >> 5;               // 32 float4 per row
            const int c4  = idx & 31;
            const float4 v = src[row * 32 + c4];
            float* dst = &xs[row * XS_STRIDE + c4 * 4];
            dst[0] = v.x; dst[1] = v.y; dst[2] = v.z; dst[3] = v.w;
        }
    }
    __syncthreads();

    v8f cs = {};
    v8f cn = {};
    const float* arow = &xs[lr * XS_STRIDE + hi * 2];

    for (int k = 0; k < D; k += 4) {
        v2f a = *(const v2f*)(arow + k);           // ds_load_b64, bank-conflict-free
        const int kk = k + hi * 2;
        v2f bs, bn;                                 // B fragments, row-major weights
        bs.x = Ws[(kk + 0) * D + n0 + lr];
        bs.y = Ws[(kk + 1) * D + n0 + lr];
        bn.x = Wn[(kk + 0) * D + n0 + lr];
        bn.y = Wn[(kk + 1) * D + n0 + lr];
        cs = __builtin_amdgcn_wmma_f32_16x16x4_f32(false, a, false, bs, (short)0, cs,
                                                   false, false);
        cn = __builtin_amdgcn_wmma_f32_16x16x4_f32(false, a, false, bn, (short)0, cn,
                                                   false, false);
    }

    const float bv = bias[n0 + lr];
    #pragma unroll
    for (int v = 0; v < 8; ++v) {
        const int m = m0 + v + hi * 8;
        const int n = n0 + lr;
        out[(long long)m * D + n] = cs[v] + bv;   // self_support + bias
        h[(long long)m * D + n]   = cn[v];        // neighbor transform
    }
}

// ------------------------------------------------------------------- scatter
// One wave per edge; lane owns 4 contiguous dwords of the 128-wide feature.
// h is L2-resident (51 MB < 192 MB), so the gather hits L2; sums via L2 atomics.
__global__ __launch_bounds__(256) void k_scatter(const long long* __restrict__ ei,
                                                 const float* __restrict__ dinv,
                                                 const float* __restrict__ h,
                                                 float* __restrict__ out) {
    const long long e = (long long)blockIdx.x * 8 + (threadIdx.x >> 5);
    if (e >= N_EDGES) return;
    const int lane = threadIdx.x & 31;
    const int r = (int)ei[e];
    const int c = (int)ei[N_EDGES + e];
    const float norm = dinv[r] * dinv[c];
    const float4 hv = *(const float4*)(h + (long long)c * D + lane * 4);
    float* o = out + (long long)r * D + lane * 4;
    atomicAdd(o + 0, norm * hv.x);
    atomicAdd(o + 1, norm * hv.y);
    atomicAdd(o + 2, norm * hv.z);
    atomicAdd(o + 3, norm * hv.w);
}

extern "C" void kernel_launch(void* const* d_in, const int* in_sizes, int n_in,
                              void* d_out, int out_size, void* d_ws, size_t ws_size,
                              hipStream_t stream) {
    const float*     x    = (const float*)d_in[0];
    const long long* ei   = (const long long*)d_in[1];   // int64 [2, E]
    const float*     Ws   = (const float*)d_in[2];
    const float*     Wn   = (const float*)d_in[3];
    const float*     bias = (const float*)d_in[4];
    float*           out  = (float*)d_out;

    float* deg  = (float*)d_ws;                 // [N]
    float* dinv = deg + N_NODES;                // [N]
    float* h    = dinv + N_NODES;               // [N, 128]

    k_zero<<<(N_NODES + 255) / 256, 256, 0, stream>>>(deg);
    k_degree<<<(N_EDGES + 255) / 256, 256, 0, stream>>>(ei, deg);
    k_rsqrt<<<(N_NODES + 255) / 256, 256, 0, stream>>>(deg, dinv);
    k_gemm_fused<<<N_NODES / 16, 256, 0, stream>>>(x, Ws, Wn, bias, out, h);
    k_scatter<<<N_EDGES / 8, 256, 0, stream>>>(ei, dinv, h, out);
}